// MeTokenPro_Model_24627342475479
// MI455X (gfx1250) — compile-verified
//
#include <hip/hip_runtime.h>
#include <hip/hip_bf16.h>

typedef __attribute__((ext_vector_type(2))) float v2f;
typedef __attribute__((ext_vector_type(8))) float v8f;

#define EMBED_DIM 512
#define NUM_TYPES 26
#define PER_TYPE 128
#define NUM_CODES (NUM_TYPES * PER_TYPE)   // 3328
#define N_TOKENS 32768
#define SAMPLED 312
#define TEMP_INV (1.0f / 0.07f)
#define NSLICE 64

// ---- workspace layout (bytes), all 16B aligned ----
// inv_norm : [0,        131072)   32768 f32
// enormsq  : [131072,   144384)   3328 f32
// se       : [144384,   783360)   312*512 f32
// counts   : [783360,   783464)   26 i32
// offsets  : [783488,   783596)   27 i32
// cursors  : [783616,   783720)   26 i32
// buckets  : [783744,   914816)   32768 i32
// idxI     : [914816,  1045888)   32768 i32
// dterm    : [1048576, 1179648)   32768 f32
// uterm    : [1179648, 1180896)   312 f32

__device__ __forceinline__ float waveAllSum(float v) {
  #pragma unroll
  for (int off = 16; off > 0; off >>= 1) v += __shfl_xor(v, off, 32);
  return v;
}

// ---------- pass 1: per-token inverse norm + type histogram ----------
__global__ void k_norm_count(const float* __restrict__ x, const int* __restrict__ Q,
                             float* __restrict__ inv_norm, int* __restrict__ counts) {
  int token = (blockIdx.x * blockDim.x + threadIdx.x) >> 5;
  int lane = threadIdx.x & 31;
  if (token >= N_TOKENS) return;
  const float* row = x + (size_t)token * EMBED_DIM;
  float s = 0.f;
  #pragma unroll
  for (int t = 0; t < 4; ++t) {
    float4 v = *(const float4*)(row + lane * 4 + t * 128);
    s += v.x * v.x + v.y * v.y + v.z * v.z + v.w * v.w;
  }
  s = waveAllSum(s);
  if (lane == 0) {
    inv_norm[token] = 1.0f / fmaxf(sqrtf(s), 1e-12f);
    atomicAdd(&counts[Q[token]], 1);
  }
}

// ---------- per-code squared norms ----------
__global__ void k_enorm(const float* __restrict__ emb, float* __restrict__ enormsq) {
  int code = (blockIdx.x * blockDim.x + threadIdx.x) >> 5;
  int lane = threadIdx.x & 31;
  if (code >= NUM_CODES) return;
  const float* row = emb + (size_t)code * EMBED_DIM;
  float s = 0.f;
  #pragma unroll
  for (int t = 0; t < 4; ++t) {
    float4 v = *(const float4*)(row + lane * 4 + t * 128);
    s += v.x * v.x + v.y * v.y + v.z * v.z + v.w * v.w;
  }
  s = waveAllSum(s);
  if (lane == 0) enormsq[code] = s;
}

__global__ void k_zero(int* __restrict__ counts) {
  if (threadIdx.x < NUM_TYPES) counts[threadIdx.x] = 0;
}

__global__ void k_scan(const int* __restrict__ counts, int* __restrict__ offsets,
                       int* __restrict__ cursors) {
  if (threadIdx.x == 0) {
    int acc = 0;
    for (int t = 0; t < NUM_TYPES; ++t) {
      offsets[t] = acc; cursors[t] = acc; acc += counts[t];
    }
    offsets[NUM_TYPES] = acc;
  }
}

__global__ void k_scatter(const int* __restrict__ Q, int* __restrict__ cursors,
                          int* __restrict__ buckets) {
  int i = blockIdx.x * blockDim.x + threadIdx.x;
  if (i < N_TOKENS) {
    int p = atomicAdd(&cursors[Q[i]], 1);
    buckets[p] = i;
  }
}

// ---------- main: per-type tiled GEMM (f32 WMMA) + per-token argmin ----------
__global__ __launch_bounds__(128) void k_argmin(
    const float* __restrict__ x, const float* __restrict__ emb,
    const float* __restrict__ inv_norm, const float* __restrict__ enormsq,
    const int* __restrict__ offsets, const int* __restrict__ buckets,
    int* __restrict__ idx_out) {
  const int type  = blockIdx.x / NSLICE;
  const int slice = blockIdx.x % NSLICE;
  const int start = offsets[type];
  const int count = offsets[type + 1] - start;
  if (count <= 0) return;
  const int ntiles = (count + 15) >> 4;

  const int tid  = threadIdx.x;
  const int lane = tid & 31;
  const int wave = tid >> 5;

  __shared__ float Ald[16][66];    // token tile, K-chunk of 64 (pad -> bank 2n)
  __shared__ float Cld[128][66];   // 128 codes, K-chunk of 64
  __shared__ float Sld[16][132];   // scores: 16 tokens x 128 codes

  const float* codeBase = emb + (size_t)type * PER_TYPE * EMBED_DIM;

  for (int tile = slice; tile < ntiles; tile += NSLICE) {
    const int tbase = start + tile * 16;
    const int ttok  = min(16, count - tile * 16);

    v8f acc0 = {};   // N-tile 2*wave
    v8f acc1 = {};   // N-tile 2*wave+1

    for (int kc = 0; kc < 8; ++kc) {        // 8 chunks of K=64
      __syncthreads();
      // stage A: 16 rows x 64 cols (8 threads/row, 8 floats each, b128 loads)
      {
        int m  = tid >> 3;
        int c0 = (tid & 7) * 8;
        int tok = buckets[tbase + min(m, ttok - 1)];
        const float* xr = x + (size_t)tok * EMBED_DIM + kc * 64;
        float inv = inv_norm[tok];
        float4 v0 = *(const float4*)(xr + c0);
        float4 v1 = *(const float4*)(xr + c0 + 4);
        Ald[m][c0 + 0] = v0.x * inv; Ald[m][c0 + 1] = v0.y * inv;
        Ald[m][c0 + 2] = v0.z * inv; Ald[m][c0 + 3] = v0.w * inv;
        Ald[m][c0 + 4] = v1.x * inv; Ald[m][c0 + 5] = v1.y * inv;
        Ald[m][c0 + 6] = v1.z * inv; Ald[m][c0 + 7] = v1.w * inv;
      }
      // stage codes: 128 rows x 64 cols (16 threads/row, float4 each, 16 passes)
      {
        int rloc = tid >> 4;
        int c0   = (tid & 15) * 4;
        #pragma unroll 4
        for (int rp = 0; rp < 16; ++rp) {
          int r = rp * 8 + rloc;
          float4 v = *(const float4*)(codeBase + (size_t)r * EMBED_DIM + kc * 64 + c0);
          Cld[r][c0 + 0] = v.x; Cld[r][c0 + 1] = v.y;
          Cld[r][c0 + 2] = v.z; Cld[r][c0 + 3] = v.w;
        }
      }
      __syncthreads();

      // 16 K-steps of V_WMMA_F32_16X16X4_F32, 2 N-tiles per wave
      const int mrow = lane & 15;
      const int ksel = (lane >> 4) << 1;   // lanes 16-31 hold K={k0+2,k0+3}
      const int n0   = wave * 32;
      #pragma unroll 4
      for (int ks = 0; ks < 16; ++ks) {
        int kk = ks * 4 + ksel;
        v2f a  = *(const v2f*)&Ald[mrow][kk];
        v2f b0 = *(const v2f*)&Cld[n0 + mrow][kk];
        v2f b1 = *(const v2f*)&Cld[n0 + 16 + mrow][kk];
        acc0 = __builtin_amdgcn_wmma_f32_16x16x4_f32(false, a, false, b0,
                                                     (short)0, acc0, false, false);
        acc1 = __builtin_amdgcn_wmma_f32_16x16x4_f32(false, a, false, b1,
                                                     (short)0, acc1, false, false);
      }
    }

    // scores: d = ||e||^2 - 2*dot  (||xn||^2 is constant per row -> irrelevant for argmin)
    {
      int nl0 = wave * 32 + (lane & 15);
      int nl1 = nl0 + 16;
      float en0 = enormsq[type * PER_TYPE + nl0];
      float en1 = enormsq[type * PER_TYPE + nl1];
      int mhi = (lane >> 4) * 8;   // D layout: VGPR r -> M=r (lanes 0-15), M=r+8 (16-31)
      #pragma unroll
      for (int r = 0; r < 8; ++r) {
        Sld[r + mhi][nl0] = en0 - 2.0f * acc0[r];
        Sld[r + mhi][nl1] = en1 - 2.0f * acc1[r];
      }
    }
    __syncthreads();
    if (tid < 16 && tid < ttok) {
      float best = Sld[tid][0]; int bi = 0;
      for (int n = 1; n < PER_TYPE; ++n) {
        float s = Sld[tid][n];
        if (s < best) { best = s; bi = n; }   // first-min, matches jnp.argmin
      }
      idx_out[buckets[tbase + tid]] = type * PER_TYPE + bi;
    }
  }
}

// ---------- output pass: quantized rows + per-token latent term + idx-as-float ----------
__global__ void k_out(const float* __restrict__ x, const float* __restrict__ emb,
                      const float* __restrict__ inv_norm, const int* __restrict__ idx,
                      float* __restrict__ out, float* __restrict__ idxF,
                      float* __restrict__ dterm) {
  int token = (blockIdx.x * blockDim.x + threadIdx.x) >> 5;
  int lane = threadIdx.x & 31;
  if (token >= N_TOKENS) return;
  int code = idx[token];
  const float* e  = emb + (size_t)code * EMBED_DIM;
  const float* xr = x + (size_t)token * EMBED_DIM;
  float xinv = inv_norm[token];
  float4 ev[4];
  float s = 0.f;
  #pragma unroll
  for (int t = 0; t < 4; ++t) {
    ev[t] = *(const float4*)(e + lane * 4 + t * 128);
    s += ev[t].x * ev[t].x + ev[t].y * ev[t].y + ev[t].z * ev[t].z + ev[t].w * ev[t].w;
  }
  s = waveAllSum(s);
  float qinv = 1.0f / fmaxf(sqrtf(s), 1e-12f);
  float d = 0.f;
  float* orow = out + (size_t)token * EMBED_DIM;
  #pragma unroll
  for (int t = 0; t < 4; ++t) {
    int c = lane * 4 + t * 128;
    float4 xv = *(const float4*)(xr + c);
    float4 q;
    q.x = ev[t].x * qinv; q.y = ev[t].y * qinv;
    q.z = ev[t].z * qinv; q.w = ev[t].w * qinv;
    float dx = q.x - xv.x * xinv, dy = q.y - xv.y * xinv;
    float dz = q.z - xv.z * xinv, dw = q.w - xv.w * xinv;
    d += dx * dx + dy * dy + dz * dz + dw * dw;
    *(float4*)(orow + c) = q;
  }
  d = waveAllSum(d);
  if (lane == 0) {
    dterm[token] = d;
    idxF[token] = (float)code;
  }
}

// ---------- uniform loss ----------
__global__ void k_uprep(const float* __restrict__ emb, const int* __restrict__ samp,
                        float* __restrict__ se) {
  int s = (blockIdx.x * blockDim.x + threadIdx.x) >> 5;
  int lane = threadIdx.x & 31;
  if (s >= SAMPLED) return;
  const float* row = emb + (size_t)samp[s] * EMBED_DIM;
  float ss = 0.f;
  float4 ev[4];
  #pragma unroll
  for (int t = 0; t < 4; ++t) {
    ev[t] = *(const float4*)(row + lane * 4 + t * 128);
    ss += ev[t].x * ev[t].x + ev[t].y * ev[t].y + ev[t].z * ev[t].z + ev[t].w * ev[t].w;
  }
  ss = waveAllSum(ss);
  float inv = 1.0f / fmaxf(sqrtf(ss), 1e-12f);
  float* orow = se + (size_t)s * EMBED_DIM;
  #pragma unroll
  for (int t = 0; t < 4; ++t) {
    int c = lane * 4 + t * 128;
    float4 q;
    q.x = ev[t].x * inv; q.y = ev[t].y * inv;
    q.z = ev[t].z * inv; q.w = ev[t].w * inv;
    *(float4*)(orow + c) = q;
  }
}

__global__ __launch_bounds__(256) void k_uloss(const float* __restrict__ se,
                                               const int* __restrict__ samp,
                                               float* __restrict__ uterm) {
  __shared__ float si[EMBED_DIM];
  __shared__ float redS[8], redP[8];
  int i = blockIdx.x;
  int tid = threadIdx.x, lane = tid & 31, wave = tid >> 5;
  for (int c = tid; c < EMBED_DIM; c += 256) si[c] = se[(size_t)i * EMBED_DIM + c];
  __syncthreads();
  int labi = samp[i] >> 7;
  float sumE = 0.f, posE = 0.f;
  for (int j = wave; j < SAMPLED; j += 8) {
    const float* rj = se + (size_t)j * EMBED_DIM;
    float dot = 0.f;
    #pragma unroll
    for (int t = 0; t < 16; ++t) {
      int c = lane + t * 32;
      dot += si[c] * rj[c];
    }
    dot = waveAllSum(dot);
    if (lane == 0 && j != i) {
      float eV = expf(dot * TEMP_INV);
      sumE += eV;
      if ((samp[j] >> 7) == labi) posE += eV;
    }
  }
  if (lane == 0) { redS[wave] = sumE; redP[wave] = posE; }
  __syncthreads();
  if (tid == 0) {
    float S = 0.f, P = 0.f;
    for (int w = 0; w < 8; ++w) { S += redS[w]; P += redP[w]; }
    uterm[i] = -logf(P / S);
  }
}

// ---------- deterministic final reduction ----------
__global__ __launch_bounds__(256) void k_final(const float* __restrict__ dterm,
                                               const float* __restrict__ uterm,
                                               float* __restrict__ lossOut) {
  __shared__ float red[256];
  int tid = threadIdx.x;
  float s = 0.f;
  for (int i = tid; i < N_TOKENS; i += 256) s += dterm[i];
  red[tid] = s;
  __syncthreads();
  for (int st = 128; st > 0; st >>= 1) {
    if (tid < st) red[tid] += red[tid + st];
    __syncthreads();
  }
  if (tid == 0) {
    lossOut[0] = 1.25f * red[0] / (float)((size_t)N_TOKENS * EMBED_DIM);
    float u = 0.f;
    for (int j = 0; j < SAMPLED; ++j) u += uterm[j];
    lossOut[1] = u / (float)SAMPLED;
  }
}

extern "C" void kernel_launch(void* const* d_in, const int* in_sizes, int n_in,
                              void* d_out, int out_size, void* d_ws, size_t ws_size,
                              hipStream_t stream) {
  const float* x    = (const float*)d_in[0];
  const float* emb  = (const float*)d_in[1];
  const int*   Q    = (const int*)d_in[2];
  const int*   samp = (const int*)d_in[3];

  char* ws = (char*)d_ws;
  float* inv_norm = (float*)(ws + 0);
  float* enormsq  = (float*)(ws + 131072);
  float* se       = (float*)(ws + 144384);
  int*   counts   = (int*)(ws + 783360);
  int*   offsets  = (int*)(ws + 783488);
  int*   cursors  = (int*)(ws + 783616);
  int*   buckets  = (int*)(ws + 783744);
  int*   idxI     = (int*)(ws + 914816);
  float* dterm    = (float*)(ws + 1048576);
  float* uterm    = (float*)(ws + 1179648);

  float* out     = (float*)d_out;
  float* lossOut = out + (size_t)N_TOKENS * EMBED_DIM;  // [loss, uloss]
  float* idxF    = lossOut + 2;

  k_zero<<<1, 32, 0, stream>>>(counts);
  k_norm_count<<<(N_TOKENS * 32) / 256, 256, 0, stream>>>(x, Q, inv_norm, counts);
  k_enorm<<<(NUM_CODES * 32) / 256, 256, 0, stream>>>(emb, enormsq);
  k_scan<<<1, 32, 0, stream>>>(counts, offsets, cursors);
  k_scatter<<<(N_TOKENS + 255) / 256, 256, 0, stream>>>(Q, cursors, buckets);
  k_argmin<<<NUM_TYPES * NSLICE, 128, 0, stream>>>(x, emb, inv_norm, enormsq,
                                                   offsets, buckets, idxI);
  k_out<<<(N_TOKENS * 32) / 256, 256, 0, stream>>>(x, emb, inv_norm, idxI,
                                                   out, idxF, dterm);
  k_uprep<<<(SAMPLED * 32 + 255) / 256, 256, 0, stream>>>(emb, samp, se);
  k_uloss<<<SAMPLED, 256, 0, stream>>>(se, samp, uterm);
  k_final<<<1, 256, 0, stream>>>(dterm, uterm, lossOut);
}